// My_Graph_to_Featuremaps3_50551765074211
// MI455X (gfx1250) — compile-verified
//
#include <hip/hip_runtime.h>

typedef __attribute__((ext_vector_type(2))) float v2f;
typedef __attribute__((ext_vector_type(4))) float v4f;
typedef __attribute__((ext_vector_type(8))) float v8f;

#define CH     256
#define NODES  7
#define HID    128
#define DHW    65536     // 16*64*64
#define BATCH  2

// ---------------------------------------------------------------------------
// Kernel 1: new_weight = inp[b] (7x128, zero-padded to 16) @ weight (128x256)
// via V_WMMA_F32_16X16X4_F32, one wave per (batch, 16-col channel tile).
// Also computes n2[b][n] = inp[b][n] . node_fea_for_hidden.
// Workspace layout (floats): [0 .. 3584) = new_weight[2][7][256]
//                            [3584 .. 3598) = n2[2][7]
// ---------------------------------------------------------------------------
__global__ __launch_bounds__(32) void prep_kernel(
    const float* __restrict__ inp,    // [2][7][128]
    const float* __restrict__ nfh,    // [128]
    const float* __restrict__ weight, // [128][256]
    float* __restrict__ ws)
{
    const int lane = threadIdx.x;
    const int b    = blockIdx.x >> 4;   // batch
    const int ct   = blockIdx.x & 15;   // 16-wide channel tile
    const int m    = lane & 15;         // M row (A) / N col (B) index
    const int hi   = lane >> 4;         // selects K offset +2 per ISA layout

    // Clamp row so loads are always in-bounds (unconditional, no EXEC churn),
    // zero the padded rows with a post-load select instead.
    const int   mc = (m < NODES) ? m : 0;
    const bool  mv = (m < NODES);
    const float* A = inp + b * NODES * HID;   // [7][128]

    v8f acc = {};
    for (int k0 = 0; k0 < HID; k0 += 4) {
        const int ka = k0 + 2 * hi;
        // A 16x4 f32 layout: VGPR0 = K=(0|2), VGPR1 = K=(1|3); lanes 0-15 M, 16-31 M w/ K+2
        const float ax = A[mc * HID + ka];
        const float ay = A[mc * HID + ka + 1];
        v2f a, bb;
        a.x = mv ? ax : 0.0f;
        a.y = mv ? ay : 0.0f;
        // B 4x16 f32 layout (mirror of A): lanes carry N, VGPRs/half-wave carry K
        bb.x = weight[(size_t)ka       * CH + ct * 16 + m];
        bb.y = weight[(size_t)(ka + 1) * CH + ct * 16 + m];
        acc = __builtin_amdgcn_wmma_f32_16x16x4_f32(
            false, a, false, bb, (short)0, acc, false, false);
    }

    // D 16x16 f32: VGPR r = rows {r, r+8}; lanes 0-15 hold M=r. Need M=0..6.
    if (hi == 0) {
        #pragma unroll
        for (int r = 0; r < NODES; ++r)
            ws[((size_t)b * NODES + r) * CH + ct * 16 + m] = acc[r];
    }

    // n2: 14 tiny dot products, one lane each (block 0 only)
    if (blockIdx.x == 0 && lane < BATCH * NODES) {
        const int b2 = lane / NODES, n = lane % NODES;
        const float* row = inp + ((size_t)b2 * NODES + n) * HID;
        float s = 0.0f;
        for (int k = 0; k < HID; ++k) s += row[k] * nfh[k];
        ws[BATCH * NODES * CH + lane] = s;
    }
}

// ---------------------------------------------------------------------------
// Kernel 2: fused streaming pass (bandwidth-bound part, ~268 MB of traffic).
//   n1[p,n]  = sum_c res[b,c,p] * W1[c,n]        (coalesced b128 NT loads)
//   attn     = softmax(n1 + n2)                  (in registers, 7 nodes)
//   out[c,p] = relu(sum_n attn[p,n]*nw[b,n,c])   (coalesced b128 NT stores)
// One thread owns 4 consecutive points. W1 / nw / n2 staged in LDS and read
// with wave-uniform addresses (bank-conflict-free broadcast). c-loops are
// unrolled x4 to keep 4 independent b128 loads/stores in flight per wave.
// ---------------------------------------------------------------------------
#define TPB 128
#define PPT 4

__device__ inline v4f vmax4(v4f a, v4f b) {
    v4f r; r.x = fmaxf(a.x, b.x); r.y = fmaxf(a.y, b.y);
           r.z = fmaxf(a.z, b.z); r.w = fmaxf(a.w, b.w); return r;
}
__device__ inline v4f vexp4(v4f a) {
    v4f r; r.x = __expf(a.x); r.y = __expf(a.y);
           r.z = __expf(a.z); r.w = __expf(a.w); return r;
}

__global__ __launch_bounds__(TPB) void fuse_kernel(
    const float* __restrict__ res,  // [2][256][65536]
    const float* __restrict__ w1,   // [256][7]
    const float* __restrict__ ws,   // new_weight + n2
    float* __restrict__ out)        // [2][256][65536]
{
    __shared__ float lds_w1[CH * NODES];   // 7 KB
    __shared__ float lds_nw[NODES * CH];   // 7 KB (this batch)
    __shared__ float lds_n2[NODES];

    const int tid  = threadIdx.x;
    const int b    = blockIdx.x >> 7;           // 128 blocks per batch
    const int pblk = blockIdx.x & 127;
    const int p0   = pblk * (TPB * PPT) + tid * PPT;

    for (int i = tid; i < CH * NODES; i += TPB) {
        lds_w1[i] = w1[i];
        lds_nw[i] = ws[(size_t)b * NODES * CH + i];
    }
    if (tid < NODES) lds_n2[tid] = ws[BATCH * NODES * CH + b * NODES + tid];
    __syncthreads();

    const float* resb = res + (size_t)b * CH * DHW + p0;
    float*       outb = out + (size_t)b * CH * DHW + p0;

    v4f acc[NODES];
    #pragma unroll
    for (int n = 0; n < NODES; ++n) {
        const float s = lds_n2[n];
        acc[n] = (v4f){s, s, s, s};
    }

    // Pass A: n1 accumulation while streaming res_feature once.
    #pragma unroll 4
    for (int c = 0; c < CH; ++c) {
        const v4f v = __builtin_nontemporal_load(
            (const v4f*)(resb + (size_t)c * DHW));
        #pragma unroll
        for (int n = 0; n < NODES; ++n)
            acc[n] += v * lds_w1[c * NODES + n];
    }

    // Softmax over the 7 nodes, per point (vector component).
    v4f mx = acc[0];
    #pragma unroll
    for (int n = 1; n < NODES; ++n) mx = vmax4(mx, acc[n]);
    v4f sum = {};
    #pragma unroll
    for (int n = 0; n < NODES; ++n) { acc[n] = vexp4(acc[n] - mx); sum += acc[n]; }
    v4f inv;
    inv.x = 1.0f / sum.x; inv.y = 1.0f / sum.y;
    inv.z = 1.0f / sum.z; inv.w = 1.0f / sum.w;
    #pragma unroll
    for (int n = 0; n < NODES; ++n) acc[n] *= inv;

    // Pass B: out = relu(attn @ new_weight), streamed out once.
    const v4f vzero = {};
    #pragma unroll 4
    for (int c = 0; c < CH; ++c) {
        v4f o = {};
        #pragma unroll
        for (int n = 0; n < NODES; ++n)
            o += acc[n] * lds_nw[n * CH + c];
        o = vmax4(o, vzero);
        __builtin_nontemporal_store(o, (v4f*)(outb + (size_t)c * DHW));
    }
}

// ---------------------------------------------------------------------------
extern "C" void kernel_launch(void* const* d_in, const int* in_sizes, int n_in,
                              void* d_out, int out_size, void* d_ws, size_t ws_size,
                              hipStream_t stream) {
    const float* inp = (const float*)d_in[0];  // [1][2][7][128]
    const float* res = (const float*)d_in[1];  // [2][256][16][64][64]
    const float* w1  = (const float*)d_in[2];  // [256][7]
    const float* nfh = (const float*)d_in[3];  // [128][1]
    const float* wgt = (const float*)d_in[4];  // [128][256]
    float* ws  = (float*)d_ws;
    float* out = (float*)d_out;

    prep_kernel<<<BATCH * 16, 32, 0, stream>>>(inp, nfh, wgt, ws);
    fuse_kernel<<<BATCH * 128, TPB, 0, stream>>>(res, w1, ws, out);
}